// FraudDetectionHybrid_65481071408448
// MI455X (gfx1250) — compile-verified
//
#include <hip/hip_runtime.h>

// ---------------------------------------------------------------------------
// Fused fraud-detection pipeline, dead-code-eliminated (AE encoder/decoder do
// not feed the output). Bandwidth-bound: 16 MB traffic -> ~0.7 us floor at
// 23.3 TB/s. 4 samples per thread: 2x global_load_b128 + 2x global_store_b128.
// All 14 nonlinearities/sample go through the gfx1250 hardware v_tanh_f32
// trans op (sigmoid rewritten as 0.5*tanh(x/2)+0.5) so the quarter-rate trans
// pipe carries 12+2 ops/sample while VOPD dual-issue carries the FMAs.
// Weight preamble is wave-uniform -> compiler scalarizes it (s_load_b512 +
// s_add_f32/s_sub_f32), costing zero VALU in the hot path.
// ---------------------------------------------------------------------------

#if defined(__has_builtin)
#if __has_builtin(__builtin_amdgcn_tanhf)
#define CDNA5_HAS_HW_TANH 1
#endif
#endif

__device__ __forceinline__ float fast_tanh(float x) {
#if defined(CDNA5_HAS_HW_TANH)
    return __builtin_amdgcn_tanhf(x);          // gfx1250 v_tanh_f32 (trans op)
#else
    // tanh(x) = 1 - 2/(exp2(2x*log2e)+1); safe at +/-inf (no inf/inf NaN)
    float t = __builtin_amdgcn_exp2f(x * 2.88539008177792681472f);
    return 1.0f - 2.0f * __builtin_amdgcn_rcpf(t + 1.0f);
#endif
}

// sigmoid(x) = 0.5*tanh(x/2) + 0.5 : 1 trans + 2 valu (vs 2 trans + 1 valu)
__device__ __forceinline__ float fast_sigmoid(float x) {
    return fmaf(0.5f, fast_tanh(0.5f * x), 0.5f);
}

struct SmallNet {
    float4 fw[4];        // fraud_W[l] = (w00,w01,w10,w11)
    float4 fb01, fb23;   // fraud biases, layers 0-1 / 2-3
    float  kc0, kc1;     // folded conv kernel: k00+k10, k01+k11
    float4 w1a, w1b;     // s_W1 rows 0-1 / 2-3 (each row is 2 floats)
    float4 b1;           // s_b1
    float4 wd;           // s_W2 row0 - row1 (softmax over 2 == sigmoid(l0-l1))
    float  bd;           // s_b2[0] - s_b2[1]
};

__device__ __forceinline__ float2 run_sample(float h0, float h1, const SmallNet& p) {
    // fraud net: 4 layers of tanh(W h + b), 2->2
    float4 w = p.fw[0];
    float n0 = fmaf(w.x, h0, fmaf(w.y, h1, p.fb01.x));
    float n1 = fmaf(w.z, h0, fmaf(w.w, h1, p.fb01.y));
    h0 = fast_tanh(n0); h1 = fast_tanh(n1);
    w = p.fw[1];
    n0 = fmaf(w.x, h0, fmaf(w.y, h1, p.fb01.z));
    n1 = fmaf(w.z, h0, fmaf(w.w, h1, p.fb01.w));
    h0 = fast_tanh(n0); h1 = fast_tanh(n1);
    w = p.fw[2];
    n0 = fmaf(w.x, h0, fmaf(w.y, h1, p.fb23.x));
    n1 = fmaf(w.z, h0, fmaf(w.w, h1, p.fb23.y));
    h0 = fast_tanh(n0); h1 = fast_tanh(n1);
    w = p.fw[3];
    n0 = fmaf(w.x, h0, fmaf(w.y, h1, p.fb23.z));
    n1 = fmaf(w.z, h0, fmaf(w.w, h1, p.fb23.w));
    h0 = fast_tanh(n0); h1 = fast_tanh(n1);

    // "quantum" conv surrogate: patch = [h0,h1;h0,h1] dot conv_k, sigmoid
    float conv = fast_sigmoid(fmaf(p.kc0, h0, p.kc1 * h1));

    // sampler QNN: inp = [h0, conv]; 2->4 tanh; 4->2 linear; softmax(2)
    float t0 = fast_tanh(fmaf(p.w1a.x, h0, fmaf(p.w1a.y, conv, p.b1.x)));
    float t1 = fast_tanh(fmaf(p.w1a.z, h0, fmaf(p.w1a.w, conv, p.b1.y)));
    float t2 = fast_tanh(fmaf(p.w1b.x, h0, fmaf(p.w1b.y, conv, p.b1.z)));
    float t3 = fast_tanh(fmaf(p.w1b.z, h0, fmaf(p.w1b.w, conv, p.b1.w)));

    float d = fmaf(p.wd.x, t0, fmaf(p.wd.y, t1, fmaf(p.wd.z, t2, fmaf(p.wd.w, t3, p.bd))));
    // softmax([l0,l1]) = [sigmoid(d), 1-sigmoid(d)], d = l0-l1
    float t = fast_tanh(0.5f * d);
    return make_float2(fmaf(0.5f, t, 0.5f), fmaf(-0.5f, t, 0.5f));
}

__global__ __launch_bounds__(256) void fraud_fused_kernel(
    const float4* __restrict__ x4,
    const float*  __restrict__ fW,   // [4,2,2]
    const float*  __restrict__ fb,   // [4,2]
    const float*  __restrict__ ck,   // [2,2]
    const float*  __restrict__ sW1,  // [4,2]
    const float*  __restrict__ sb1,  // [4]
    const float*  __restrict__ sW2,  // [2,4]
    const float*  __restrict__ sb2,  // [2]
    float4* __restrict__ out4,
    int npair)                       // number of float4-PAIRS (8 floats / 4 samples)
{
    int i = blockIdx.x * blockDim.x + threadIdx.x;
    if (i >= npair) return;

    SmallNet p;
    const float4* fW4 = (const float4*)fW;
    p.fw[0] = fW4[0]; p.fw[1] = fW4[1]; p.fw[2] = fW4[2]; p.fw[3] = fW4[3];
    const float4* fb4 = (const float4*)fb;
    p.fb01 = fb4[0]; p.fb23 = fb4[1];
    float4 k = ((const float4*)ck)[0];
    p.kc0 = k.x + k.z;                 // conv_k[0][0] + conv_k[1][0]
    p.kc1 = k.y + k.w;                 // conv_k[0][1] + conv_k[1][1]
    const float4* w14 = (const float4*)sW1;
    p.w1a = w14[0]; p.w1b = w14[1];
    p.b1 = ((const float4*)sb1)[0];
    const float4* w24 = (const float4*)sW2;
    float4 r0 = w24[0], r1 = w24[1];
    p.wd = make_float4(r0.x - r1.x, r0.y - r1.y, r0.z - r1.z, r0.w - r1.w);
    p.bd = sb2[0] - sb2[1];

    float4 va = x4[2 * i + 0];         // samples 0,1
    float4 vb = x4[2 * i + 1];         // samples 2,3
    float2 q0 = run_sample(va.x, va.y, p);
    float2 q1 = run_sample(va.z, va.w, p);
    float2 q2 = run_sample(vb.x, vb.y, p);
    float2 q3 = run_sample(vb.z, vb.w, p);
    out4[2 * i + 0] = make_float4(q0.x, q0.y, q1.x, q1.y);   // [p0,p1] row-major [B,2]
    out4[2 * i + 1] = make_float4(q2.x, q2.y, q3.x, q3.y);
}

extern "C" void kernel_launch(void* const* d_in, const int* in_sizes, int n_in,
                              void* d_out, int out_size, void* d_ws, size_t ws_size,
                              hipStream_t stream) {
    (void)n_in; (void)out_size; (void)d_ws; (void)ws_size;
    const float4* x4  = (const float4*)d_in[0];
    const float*  fW  = (const float*)d_in[1];
    const float*  fb  = (const float*)d_in[2];
    // d_in[3..14] = autoencoder weights: dead code, never read.
    const float*  ck  = (const float*)d_in[15];
    const float*  sW1 = (const float*)d_in[16];
    const float*  sb1 = (const float*)d_in[17];
    const float*  sW2 = (const float*)d_in[18];
    const float*  sb2 = (const float*)d_in[19];

    int npair  = in_sizes[0] / 8;      // B*2 floats -> B/4 chunks of 4 samples
    int blocks = (npair + 255) / 256;  // B=1048576 -> 1024 blocks x 256 threads
    fraud_fused_kernel<<<blocks, 256, 0, stream>>>(x4, fW, fb, ck, sW1, sb1, sW2, sb2,
                                                   (float4*)d_out, npair);
}